// GraphAttentionLayer_2018634629223
// MI455X (gfx1250) — compile-verified
//
#include <hip/hip_runtime.h>
#include <math.h>

typedef float v2f __attribute__((ext_vector_type(2)));
typedef float v8f __attribute__((ext_vector_type(8)));

#define N_NODES 4096
#define IN_FEAT 256
#define OUT_FEAT 256
#define NHEAD 4
#define HDIM 64
#define MASK_WPR (N_NODES / 32)   // 128 words per mask row
#define NEG_SLOPE 0.2f
#define SM_SCALE 0.125f           // 1/sqrt(64)

static __device__ __forceinline__ v8f wmma_f32(v2f a, v2f b, v8f c) {
    // V_WMMA_F32_16X16X4_F32: D = A(16x4,f32) * B(4x16,f32) + C(16x16,f32)
    return __builtin_amdgcn_wmma_f32_16x16x4_f32(false, a, false, b, (short)0, c,
                                                 false, false);
}

// ---------------------------------------------------------------- mask build
__global__ __launch_bounds__(256) void zero_mask_kernel(unsigned* __restrict__ mask) {
    int stride = gridDim.x * blockDim.x;
    for (int i = blockIdx.x * blockDim.x + threadIdx.x; i < N_NODES * MASK_WPR; i += stride)
        mask[i] = 0u;
}

__global__ __launch_bounds__(256) void scatter_edges_kernel(const int* __restrict__ ei,
                                                            int E,
                                                            unsigned* __restrict__ mask) {
    int e = blockIdx.x * blockDim.x + threadIdx.x;
    if (e >= E) return;
    int n = ei[e];        // src -> score row
    int m = ei[E + e];    // dst -> score col
    atomicOr(&mask[n * MASK_WPR + (m >> 5)], 1u << (m & 31));
}

// ------------------------------------------------------------ QKV projection
// One wave per 16x16 tile of one of Q/K/V. GEMM: out[n,o] = x[n,:] . W[o,:] + b[o]
// Output layout: [H][N][HDIM] per matrix.
__global__ __launch_bounds__(256) void qkv_proj_kernel(
    const float* __restrict__ x,
    const float* __restrict__ Wq, const float* __restrict__ bq,
    const float* __restrict__ Wk, const float* __restrict__ bk,
    const float* __restrict__ Wv, const float* __restrict__ bv,
    float* __restrict__ Qo, float* __restrict__ Ko, float* __restrict__ Vo) {
    int wave = (blockIdx.x * blockDim.x + threadIdx.x) >> 5;
    int lane = threadIdx.x & 31;
    int half = lane >> 4;
    int lq   = lane & 15;
    int mat  = wave / 4096;              // 0=Q 1=K 2=V
    int t    = wave - mat * 4096;
    int n0   = (t >> 4) << 4;            // node-row tile
    int o0   = (t & 15) << 4;            // out-feature tile
    const float* W = (mat == 0) ? Wq : (mat == 1) ? Wk : Wv;
    const float* b = (mat == 0) ? bq : (mat == 1) ? bk : bv;
    float*       O = (mat == 0) ? Qo : (mat == 1) ? Ko : Vo;

    const float* arow = x + (size_t)(n0 + lq) * IN_FEAT;   // A: row = lq (both halves)
    const float* brow = W + (size_t)(o0 + lq) * IN_FEAT;   // B col n = lq -> W row o0+lq

    v8f acc = {0.f, 0.f, 0.f, 0.f, 0.f, 0.f, 0.f, 0.f};
#pragma unroll
    for (int k0 = 0; k0 < IN_FEAT; k0 += 4) {
        v2f a, bb;
        a.x  = arow[k0 + 2 * half + 0];
        a.y  = arow[k0 + 2 * half + 1];
        bb.x = brow[k0 + 2 * half + 0];
        bb.y = brow[k0 + 2 * half + 1];
        acc = wmma_f32(a, bb, acc);
    }

    int col  = o0 + lq;
    int h    = col >> 6;
    int d    = col & 63;
    float bs = b[col];
    float* outp = O + ((size_t)h * N_NODES + n0) * HDIM + d;
#pragma unroll
    for (int v = 0; v < 8; ++v) {
        int row = v + 8 * half;          // C/D layout: M = v + 8*half, N = lq
        outp[(size_t)row * HDIM] = acc[v] + bs;
        // advance handled via row index
    }
}

// --------------------------------------------------------- fused attention
// One wave = 16 query rows of one head. Online softmax over 256 chunks of 16 keys.
__global__ __launch_bounds__(128) void attn_kernel(
    const float* __restrict__ Q, const float* __restrict__ K,
    const float* __restrict__ V, const unsigned* __restrict__ mask,
    float* __restrict__ agg) {
    __shared__ float smem[4 * 288];          // per wave: P[16][17] + stats[16]
    int wid  = threadIdx.x >> 5;
    int lane = threadIdx.x & 31;
    int half = lane >> 4;
    int lq   = lane & 15;
    float* P  = smem + wid * 288;
    float* st = P + 272;

    int job = blockIdx.x * 4 + wid;          // H * (N/16) = 1024 jobs
    int h   = job >> 8;
    int n0  = (job & 255) << 4;
    const float* Qh = Q + (size_t)h * N_NODES * HDIM;
    const float* Kh = K + (size_t)h * N_NODES * HDIM;
    const float* Vh = V + (size_t)h * N_NODES * HDIM;

    // Preload Q as 16 A-fragments (K-dim = 64, steps of 4)
    v2f aQ[16];
    const float* qrow = Qh + (size_t)(n0 + lq) * HDIM;
#pragma unroll
    for (int kk = 0; kk < 16; ++kk) {
        aQ[kk].x = qrow[4 * kk + 2 * half + 0];
        aQ[kk].y = qrow[4 * kk + 2 * half + 1];
    }

    v8f O0 = {0.f,0.f,0.f,0.f,0.f,0.f,0.f,0.f};
    v8f O1 = O0, O2 = O0, O3 = O0;           // d-tiles 0..3 (16 cols each)
    float m_r = -INFINITY, l_r = 0.f;        // row state, valid in lanes 0..15

    for (int j = 0; j < N_NODES / 16; ++j) {
        // ---- S = Q . K^T (16x16, K-dim 64) ----
        v8f cS = {0.f,0.f,0.f,0.f,0.f,0.f,0.f,0.f};
        const float* krow = Kh + (size_t)(j * 16 + lq) * HDIM;  // B col = key lq
#pragma unroll
        for (int kk = 0; kk < 16; ++kk) {
            v2f bK;
            bK.x = krow[4 * kk + 2 * half + 0];
            bK.y = krow[4 * kk + 2 * half + 1];
            cS = wmma_f32(aQ[kk], bK, cS);
        }
        __syncthreads();  // previous iteration's P fully consumed
        // ---- scale + LeakyReLU + mask, stage into LDS ----
#pragma unroll
        for (int v = 0; v < 8; ++v) {
            int row  = v + 8 * half;
            float s  = cS[v] * SM_SCALE;
            s = (s >= 0.f) ? s : NEG_SLOPE * s;
            unsigned w = mask[(size_t)(n0 + row) * MASK_WPR + (j >> 1)];
            int bit = ((j & 1) << 4) + lq;
            if (!((w >> bit) & 1u)) s = -INFINITY;
            P[row * 17 + lq] = s;
        }
        __syncthreads();
        // ---- online softmax, row r handled by lane r (0..15) ----
        if (lane < 16) {
            int row = lq;
            float cmax = -INFINITY;
#pragma unroll
            for (int c = 0; c < 16; ++c) cmax = fmaxf(cmax, P[row * 17 + c]);
            float nm = fmaxf(m_r, cmax);
            float corr, rs = 0.f;
            if (nm == -INFINITY) {           // no valid key seen yet
                corr = 1.f;
#pragma unroll
                for (int c = 0; c < 16; ++c) P[row * 17 + c] = 0.f;
            } else {
                corr = __expf(m_r - nm);     // m_r=-inf -> 0
#pragma unroll
                for (int c = 0; c < 16; ++c) {
                    float p = __expf(P[row * 17 + c] - nm);
                    P[row * 17 + c] = p;
                    rs += p;
                }
                m_r = nm;
            }
            l_r = l_r * corr + rs;
            st[row] = corr;
        }
        __syncthreads();
        // ---- rescale O, then O += P . V ----
#pragma unroll
        for (int v = 0; v < 8; ++v) {
            float corr = st[v + 8 * half];
            O0[v] *= corr; O1[v] *= corr; O2[v] *= corr; O3[v] *= corr;
        }
        v2f aP[4];
#pragma unroll
        for (int kk = 0; kk < 4; ++kk) {     // A-frag of P: row = lq, k = 4kk+2*half+{0,1}
            aP[kk].x = P[lq * 17 + 4 * kk + 2 * half + 0];
            aP[kk].y = P[lq * 17 + 4 * kk + 2 * half + 1];
        }
#pragma unroll
        for (int kk = 0; kk < 4; ++kk) {
            const float* vb = Vh + (size_t)(j * 16 + 4 * kk + 2 * half) * HDIM + lq;
            v2f b0, b1, b2, b3;
            b0.x = vb[0];  b0.y = vb[HDIM];
            b1.x = vb[16]; b1.y = vb[HDIM + 16];
            b2.x = vb[32]; b2.y = vb[HDIM + 32];
            b3.x = vb[48]; b3.y = vb[HDIM + 48];
            O0 = wmma_f32(aP[kk], b0, O0);
            O1 = wmma_f32(aP[kk], b1, O1);
            O2 = wmma_f32(aP[kk], b2, O2);
            O3 = wmma_f32(aP[kk], b3, O3);
        }
    }
    // ---- normalize and write agg in [N][OUT_FEAT] layout (head-concat) ----
    __syncthreads();
    if (lane < 16) st[lq] = 1.f / l_r;
    __syncthreads();
    float* outp = agg + (size_t)n0 * OUT_FEAT + h * HDIM;
#pragma unroll
    for (int v = 0; v < 8; ++v) {
        int row  = v + 8 * half;
        float iv = st[row];
        outp[(size_t)row * OUT_FEAT + 0  + lq] = O0[v] * iv;
        outp[(size_t)row * OUT_FEAT + 16 + lq] = O1[v] * iv;
        outp[(size_t)row * OUT_FEAT + 32 + lq] = O2[v] * iv;
        outp[(size_t)row * OUT_FEAT + 48 + lq] = O3[v] * iv;
    }
}

// --------------------------------------------------------- output projection
__global__ __launch_bounds__(256) void out_proj_kernel(
    const float* __restrict__ A, const float* __restrict__ Wo,
    const float* __restrict__ bo, float* __restrict__ out) {
    int wave = (blockIdx.x * blockDim.x + threadIdx.x) >> 5;
    int lane = threadIdx.x & 31;
    int half = lane >> 4;
    int lq   = lane & 15;
    int n0   = (wave >> 4) << 4;
    int o0   = (wave & 15) << 4;

    const float* arow = A + (size_t)(n0 + lq) * OUT_FEAT;
    const float* brow = Wo + (size_t)(o0 + lq) * OUT_FEAT;
    v8f acc = {0.f,0.f,0.f,0.f,0.f,0.f,0.f,0.f};
#pragma unroll
    for (int k0 = 0; k0 < OUT_FEAT; k0 += 4) {
        v2f a, bb;
        a.x  = arow[k0 + 2 * half + 0];
        a.y  = arow[k0 + 2 * half + 1];
        bb.x = brow[k0 + 2 * half + 0];
        bb.y = brow[k0 + 2 * half + 1];
        acc = wmma_f32(a, bb, acc);
    }
    float bs = bo[o0 + lq];
#pragma unroll
    for (int v = 0; v < 8; ++v) {
        int row = v + 8 * half;
        out[(size_t)(n0 + row) * OUT_FEAT + o0 + lq] = acc[v] + bs;
    }
}

extern "C" void kernel_launch(void* const* d_in, const int* in_sizes, int n_in,
                              void* d_out, int out_size, void* d_ws, size_t ws_size,
                              hipStream_t stream) {
    const float* x  = (const float*)d_in[0];
    const int*   ei = (const int*)d_in[1];
    const float* Wq = (const float*)d_in[2];
    const float* bq = (const float*)d_in[3];
    const float* Wk = (const float*)d_in[4];
    const float* bk = (const float*)d_in[5];
    const float* Wv = (const float*)d_in[6];
    const float* bv = (const float*)d_in[7];
    const float* Wo = (const float*)d_in[8];
    const float* bo = (const float*)d_in[9];
    int E = in_sizes[1] / 2;

    char* ws = (char*)d_ws;
    unsigned* mask = (unsigned*)ws;                      //  2 MB bitmask
    float* Q   = (float*)(ws + (size_t)(2  << 20));      //  4 MB [H][N][64]
    float* K   = (float*)(ws + (size_t)(6  << 20));      //  4 MB
    float* V   = (float*)(ws + (size_t)(10 << 20));      //  4 MB
    float* agg = (float*)(ws + (size_t)(14 << 20));      //  4 MB [N][256]

    zero_mask_kernel<<<512, 256, 0, stream>>>(mask);
    scatter_edges_kernel<<<(E + 255) / 256, 256, 0, stream>>>(ei, E, mask);
    qkv_proj_kernel<<<1536, 256, 0, stream>>>(x, Wq, bq, Wk, bk, Wv, bv, Q, K, V);
    attn_kernel<<<256, 128, 0, stream>>>(Q, K, V, mask, agg);
    out_proj_kernel<<<512, 256, 0, stream>>>(agg, Wo, bo, (float*)d_out);
}